// TransformerDecoderLayer_equi_v3_21603685499624
// MI455X (gfx1250) — compile-verified
//
#include <hip/hip_runtime.h>
#include <hip/hip_bf16.h>
#include <math.h>

// Problem constants (from reference)
#define NN 1024
#define KK 16
#define DD 512
#define HH 8
#define FFF 2048
#define MULA 64
#define ESS 64
#define EE (NN*KK)      // 16384
#define HD (DD/HH)      // 64

typedef __attribute__((ext_vector_type(2))) float v2f;
typedef __attribute__((ext_vector_type(8))) float v8f;

#define WMMA_F32(acc, af, bf) \
  acc = __builtin_amdgcn_wmma_f32_16x16x4_f32(false, af, false, bf, (short)0, acc, false, false)

// ---------------------------------------------------------------------------
// Generic fp32 WMMA GEMM:  C[M x Nout] = A[M x Kd] * W[Nout x Kd]^T
// One wave computes a 64x64 tile via 4x4 blocking of v_wmma_f32_16x16x4_f32.
// Ping-pong double-buffered over K (step 8 = two WMMA K-steps): fragment
// loads for one buffer overlap the 16 WMMAs of the other, with no register
// rotation copies. Requires (Kd/4) even — true for all call sites (512/2048).
// ---------------------------------------------------------------------------
__global__ __launch_bounds__(256) void gemm_wmma_f32(
    const float* __restrict__ A, const float* __restrict__ W,
    float* __restrict__ C, int M, int Nout, int Kd)
{
  const int lane = threadIdx.x & 31;
  const int wave = threadIdx.x >> 5;
  const int tiles_n = Nout >> 6;
  const int tiles_m = M >> 6;
  int tile = blockIdx.x * 8 + wave;          // wave-uniform -> EXEC stays full
  if (tile >= tiles_m * tiles_n) return;
  const int m0 = (tile / tiles_n) * 64;
  const int n0 = (tile % tiles_n) * 64;

  const int r  = lane & 15;          // row-in-fragment
  const int kk = (lane >> 4) * 2;    // K sub-offset: lanes 0-15 -> {0,1}, 16-31 -> {2,3}

  const float* Ap[4];
  const float* Wp[4];
#pragma unroll
  for (int i = 0; i < 4; ++i) {
    Ap[i] = A + (size_t)(m0 + 16 * i + r) * Kd + kk;
    Wp[i] = W + (size_t)(n0 + 16 * i + r) * Kd + kk;
  }

  v8f acc[4][4];
#pragma unroll
  for (int i = 0; i < 4; ++i)
#pragma unroll
    for (int j = 0; j < 4; ++j) acc[i][j] = (v8f){};

  v2f a0[4], b0[4], a1[4], b1[4];
#pragma unroll
  for (int i = 0; i < 4; ++i) { a0[i] = *(const v2f*)(Ap[i]); b0[i] = *(const v2f*)(Wp[i]); }

#pragma unroll 1
  for (int k = 4; k < Kd; k += 8) {
    // load odd step into buffer 1
#pragma unroll
    for (int i = 0; i < 4; ++i) { a1[i] = *(const v2f*)(Ap[i] + k); b1[i] = *(const v2f*)(Wp[i] + k); }
    // compute even step (buffer 0)
#pragma unroll
    for (int i = 0; i < 4; ++i)
#pragma unroll
      for (int j = 0; j < 4; ++j) WMMA_F32(acc[i][j], a0[i], b0[j]);
    // load next even step into buffer 0 (wave-uniform guard)
    if (k + 4 < Kd) {
#pragma unroll
      for (int i = 0; i < 4; ++i) { a0[i] = *(const v2f*)(Ap[i] + k + 4); b0[i] = *(const v2f*)(Wp[i] + k + 4); }
    }
    // compute odd step (buffer 1)
#pragma unroll
    for (int i = 0; i < 4; ++i)
#pragma unroll
      for (int j = 0; j < 4; ++j) WMMA_F32(acc[i][j], a1[i], b1[j]);
  }

  const int cn    = lane & 15;
  const int rbase = (lane >> 4) * 8;
#pragma unroll
  for (int i = 0; i < 4; ++i)
#pragma unroll
    for (int j = 0; j < 4; ++j)
#pragma unroll
      for (int q = 0; q < 8; ++q)
        C[(size_t)(m0 + 16 * i + rbase + q) * Nout + n0 + 16 * j + cn] = acc[i][j][q];
}

// ---------------------------------------------------------------------------
// Deinterleave (R, D, 3) -> 3 planar R*D matrices
// ---------------------------------------------------------------------------
__global__ void pack3_kernel(const float* __restrict__ src, float* __restrict__ dst, int R)
{
  size_t idx = (size_t)blockIdx.x * blockDim.x + threadIdx.x;
  size_t total = (size_t)R * DD;
  if (idx >= total) return;
  size_t r = idx / DD, c = idx % DD;
  const float* s = src + r * (3 * DD) + c * 3;
  dst[idx]             = s[0];
  dst[total + idx]     = s[1];
  dst[2 * total + idx] = s[2];
}

__global__ void transpose_walpha(const float* __restrict__ W, float* __restrict__ WT)
{
  int idx = blockIdx.x * blockDim.x + threadIdx.x;  // m*D + c
  if (idx >= MULA * DD) return;
  int m = idx / DD, c = idx % DD;
  WT[(size_t)m * DD + c] = W[(size_t)c * MULA + m];
}

// msg = mv + tv (broadcast over k), in-place on mv planes
__global__ void msg_add_kernel(float* __restrict__ mv_p, const float* __restrict__ tv_p)
{
  size_t idx = (size_t)blockIdx.x * blockDim.x + threadIdx.x;
  const size_t plane = (size_t)EE * DD;
  if (idx >= plane) return;
  size_t e = idx / DD, c = idx % DD;
  size_t tidx = (e / KK) * DD + c;
  const size_t tplane = (size_t)NN * DD;
  mv_p[idx]             += tv_p[tidx];
  mv_p[plane + idx]     += tv_p[tplane + tidx];
  mv_p[2 * plane + idx] += tv_p[2 * tplane + tidx];
}

// ---------------------------------------------------------------------------
// Per-edge: 3-layer MLP (silu,silu,linear) on edge_scalars -> w[1536] in LDS,
// then p_sc and p_v (p_v overwrites msg planes in place).
// ---------------------------------------------------------------------------
__global__ __launch_bounds__(256) void edge_kernel(
    const float* __restrict__ es, const float* __restrict__ sh,
    const float* __restrict__ fw1, const float* __restrict__ fb1,
    const float* __restrict__ fw2, const float* __restrict__ fb2,
    const float* __restrict__ fw3, const float* __restrict__ fb3,
    float* __restrict__ msg_p, float* __restrict__ p_sc)
{
  __shared__ float s_es[ESS];
  __shared__ float s_h1[32];
  __shared__ float s_h2[32];
  __shared__ float s_w[3 * DD];
  int e = blockIdx.x, t = threadIdx.x;

  if (t < ESS) s_es[t] = es[(size_t)e * ESS + t];
  __syncthreads();
  if (t < 32) {
    float acc = fb1[t];
    for (int i = 0; i < ESS; ++i) acc += s_es[i] * fw1[i * 32 + t];
    s_h1[t] = acc / (1.f + expf(-acc));         // silu
  }
  __syncthreads();
  if (t < 32) {
    float acc = fb2[t];
    for (int i = 0; i < 32; ++i) acc += s_h1[i] * fw2[i * 32 + t];
    s_h2[t] = acc / (1.f + expf(-acc));
  }
  __syncthreads();
  for (int j = t; j < 3 * DD; j += 256) {
    float acc = fb3[j];
    for (int i = 0; i < 32; ++i) acc += s_h2[i] * fw3[i * (3 * DD) + j];
    s_w[j] = acc;
  }
  __syncthreads();

  const float s  = sh[(size_t)e * 4 + 0];
  const float v0 = sh[(size_t)e * 4 + 1];
  const float v1 = sh[(size_t)e * 4 + 2];
  const float v2 = sh[(size_t)e * 4 + 3];
  const float inv_s3 = 0.5773502691896258f;  // 1/sqrt(3)
  const float inv_s2 = 0.7071067811865476f;  // 1/sqrt(2)
  const size_t plane = (size_t)EE * DD;

  for (int c = t; c < DD; c += 256) {
    size_t idx = (size_t)e * DD + c;
    float m0 = msg_p[idx];
    float m1 = msg_p[plane + idx];
    float m2 = msg_p[2 * plane + idx];
    float w0 = s_w[c], w1 = s_w[DD + c], w2 = s_w[2 * DD + c];
    p_sc[idx] = w0 * (m0 * v0 + m1 * v1 + m2 * v2) * inv_s3;
    float cx = m1 * v2 - m2 * v1;
    float cy = m2 * v0 - m0 * v2;
    float cz = m0 * v1 - m1 * v0;
    msg_p[idx]             = w1 * m0 * s + w2 * cx * inv_s2;
    msg_p[plane + idx]     = w1 * m1 * s + w2 * cy * inv_s2;
    msg_p[2 * plane + idx] = w1 * m2 * s + w2 * cz * inv_s2;
  }
}

// ---------------------------------------------------------------------------
// LayerNorm(64) + smooth_lrelu + per-head dot with alpha_dot -> logits[E,8]
// ---------------------------------------------------------------------------
__global__ __launch_bounds__(64) void alpha_kernel(
    const float* __restrict__ a_pre, const float* __restrict__ g,
    const float* __restrict__ b, const float* __restrict__ adot,
    float* __restrict__ logits)
{
  __shared__ float red[64];
  __shared__ float vals[64];
  int e = blockIdx.x, t = threadIdx.x;
  float x = a_pre[(size_t)e * 64 + t];
  red[t] = x; __syncthreads();
  for (int s = 32; s > 0; s >>= 1) { if (t < s) red[t] += red[t + s]; __syncthreads(); }
  float mu = red[0] * (1.f / 64.f); __syncthreads();
  red[t] = (x - mu) * (x - mu); __syncthreads();
  for (int s = 32; s > 0; s >>= 1) { if (t < s) red[t] += red[t + s]; __syncthreads(); }
  float var = red[0] * (1.f / 64.f);
  float ln = (x - mu) * rsqrtf(var + 1e-6f) * g[t] + b[t];
  float y = 0.6f * ln + 0.4f * ln * tanhf(0.5f * ln);  // smooth_lrelu(a=0.2)
  vals[t] = y; __syncthreads();
  if (t < HH) {
    float acc = 0.f;
    for (int m = 0; m < 8; ++m) acc += vals[t * 8 + m] * adot[t * 8 + m];
    logits[(size_t)e * HH + t] = acc;
  }
}

__global__ void softmax_kernel(const float* __restrict__ logits, float* __restrict__ aw)
{
  int idx = blockIdx.x * blockDim.x + threadIdx.x;  // n*H + h
  if (idx >= NN * HH) return;
  int n = idx / HH, h = idx % HH;
  float mx = -1e30f;
  for (int k = 0; k < KK; ++k) mx = fmaxf(mx, logits[((size_t)n * KK + k) * HH + h]);
  float vals[KK], sum = 0.f;
  for (int k = 0; k < KK; ++k) { vals[k] = expf(logits[((size_t)n * KK + k) * HH + h] - mx); sum += vals[k]; }
  float inv = 1.f / sum;
  for (int k = 0; k < KK; ++k) aw[((size_t)n * KK + k) * HH + h] = vals[k] * inv;
}

// out[n, o, x] = sum_k aw[n,k, o/HD] * val[(n,k), o, x]
__global__ void attn_combine(const float* __restrict__ aw, const float* __restrict__ val_p,
                             float* __restrict__ out_p)
{
  int idx = blockIdx.x * blockDim.x + threadIdx.x;  // n*D + c
  if (idx >= NN * DD) return;
  int n = idx / DD, c = idx % DD;
  int h = c >> 6;                                   // c / HD
  const size_t plane = (size_t)EE * DD;
  const size_t oplane = (size_t)NN * DD;
  float o0 = 0.f, o1 = 0.f, o2 = 0.f;
  for (int k = 0; k < KK; ++k) {
    float w = aw[((size_t)n * KK + k) * HH + h];
    size_t vidx = ((size_t)n * KK + k) * DD + c;
    o0 += w * val_p[vidx];
    o1 += w * val_p[plane + vidx];
    o2 += w * val_p[2 * plane + vidx];
  }
  out_p[idx] = o0; out_p[oplane + idx] = o1; out_p[2 * oplane + idx] = o2;
}

// ---------------------------------------------------------------------------
// vn_ln(A + B) with gamma/beta; out planar (interleaved==0) or (N, D*3)
// ---------------------------------------------------------------------------
__global__ __launch_bounds__(256) void vnln_kernel(
    const float* __restrict__ A_p, const float* __restrict__ B_p,
    const float* __restrict__ g, const float* __restrict__ b,
    float* __restrict__ out, int interleaved)
{
  __shared__ float red[256];
  int n = blockIdx.x, t = threadIdx.x;
  const size_t plane = (size_t)NN * DD;
  float x0[2], x1[2], x2[2], nrm[2];
  float lsum = 0.f, lsq = 0.f;
  for (int i = 0; i < 2; ++i) {
    int c = t + i * 256;
    size_t idx = (size_t)n * DD + c;
    float a0 = A_p[idx] + B_p[idx];
    float a1 = A_p[plane + idx] + B_p[plane + idx];
    float a2 = A_p[2 * plane + idx] + B_p[2 * plane + idx];
    x0[i] = a0; x1[i] = a1; x2[i] = a2;
    float nn = sqrtf(a0 * a0 + a1 * a1 + a2 * a2 + 1e-6f);
    nrm[i] = nn; lsum += nn; lsq += nn * nn;
  }
  red[t] = lsum; __syncthreads();
  for (int s = 128; s > 0; s >>= 1) { if (t < s) red[t] += red[t + s]; __syncthreads(); }
  float mu = red[0] * (1.f / DD); __syncthreads();
  red[t] = lsq; __syncthreads();
  for (int s = 128; s > 0; s >>= 1) { if (t < s) red[t] += red[t + s]; __syncthreads(); }
  float var = red[0] * (1.f / DD) - mu * mu;
  float inv_sd = rsqrtf(var + 1e-6f);
  for (int i = 0; i < 2; ++i) {
    int c = t + i * 256;
    float ln = (nrm[i] - mu) * inv_sd * g[c] + b[c];
    float scale = ln / nrm[i];
    if (interleaved) {
      float* o = out + (size_t)n * (3 * DD) + (size_t)c * 3;
      o[0] = x0[i] * scale; o[1] = x1[i] * scale; o[2] = x2[i] * scale;
    } else {
      size_t idx = (size_t)n * DD + c;
      out[idx] = x0[i] * scale;
      out[plane + idx] = x1[i] * scale;
      out[2 * plane + idx] = x2[i] * scale;
    }
  }
}

// Orthogonal-projection gate on (y, dvec), in place on y planes
__global__ void gate_kernel(float* __restrict__ y_p, const float* __restrict__ dv_p)
{
  size_t idx = (size_t)blockIdx.x * blockDim.x + threadIdx.x;  // n*FF + f
  const size_t plane = (size_t)NN * FFF;
  if (idx >= plane) return;
  float y0 = y_p[idx], y1 = y_p[plane + idx], y2 = y_p[2 * plane + idx];
  float d0 = dv_p[idx], d1 = dv_p[plane + idx], d2 = dv_p[2 * plane + idx];
  float dot = y0 * d0 + y1 * d1 + y2 * d2;
  if (dot < 0.f) {
    float f = dot / (d0 * d0 + d1 * d1 + d2 * d2 + 1e-6f);
    float p0 = y0 - f * d0, p1 = y1 - f * d1, p2 = y2 - f * d2;
    y_p[idx]             = 0.2f * y0 + 0.8f * p0;
    y_p[plane + idx]     = 0.2f * y1 + 0.8f * p1;
    y_p[2 * plane + idx] = 0.2f * y2 + 0.8f * p2;
  }
}

// ---------------------------------------------------------------------------
static inline int gemm_grid(int M, int Nout) { return ((M / 64) * (Nout / 64) + 7) / 8; }

extern "C" void kernel_launch(void* const* d_in, const int* in_sizes, int n_in,
                              void* d_out, int out_size, void* d_ws, size_t ws_size,
                              hipStream_t stream)
{
  const float* tgt      = (const float*)d_in[0];
  const float* memory   = (const float*)d_in[1];
  const float* edge_sh  = (const float*)d_in[2];
  const float* edge_sc  = (const float*)d_in[3];
  const float* W_src    = (const float*)d_in[4];
  const float* W_dst    = (const float*)d_in[5];
  const float* fc_w1    = (const float*)d_in[6];
  const float* fc_b1    = (const float*)d_in[7];
  const float* fc_w2    = (const float*)d_in[8];
  const float* fc_b2    = (const float*)d_in[9];
  const float* fc_w3    = (const float*)d_in[10];
  const float* fc_b3    = (const float*)d_in[11];
  const float* W_alpha  = (const float*)d_in[12];
  const float* ln_a_g   = (const float*)d_in[13];
  const float* ln_a_b   = (const float*)d_in[14];
  const float* a_dot    = (const float*)d_in[15];
  const float* W_value  = (const float*)d_in[16];
  const float* W_proj   = (const float*)d_in[17];
  const float* n1_g     = (const float*)d_in[18];
  const float* n1_b     = (const float*)d_in[19];
  const float* n2_g     = (const float*)d_in[20];
  const float* n2_b     = (const float*)d_in[21];
  const float* W_ff1    = (const float*)d_in[22];
  const float* W_ffd    = (const float*)d_in[23];
  const float* W_ff2    = (const float*)d_in[24];
  float* out = (float*)d_out;

  // Workspace carve-up (floats)
  float* ws = (float*)d_ws;
  size_t off = 0;
  const size_t ND3 = (size_t)3 * NN * DD;
  const size_t ED3 = (size_t)3 * EE * DD;
  float* T_p    = ws + off; off += ND3;                 // packed tgt
  float* tv_p   = ws + off; off += ND3;
  float* Mem_p  = ws + off; off += ED3;                 // packed memory, reused for val
  float* B1     = ws + off; off += ED3;                 // mv -> msg -> p_v
  float* p_sc   = ws + off; off += (size_t)EE * DD;
  float* WaT    = ws + off; off += (size_t)MULA * DD;
  float* a_pre  = ws + off; off += (size_t)EE * MULA;
  float* logit  = ws + off; off += (size_t)EE * HH;
  float* aw     = ws + off; off += (size_t)EE * HH;
  float* out_p  = ws + off; off += ND3;
  float* proj_p = ws + off; off += ND3;
  float* x1_p   = ws + off; off += ND3;
  float* y_p    = ws + off; off += (size_t)3 * NN * FFF;
  float* dv_p   = ws + off; off += (size_t)3 * NN * FFF;
  float* y2_p   = ws + off; off += ND3;
  (void)ws_size; (void)in_sizes; (void)n_in; (void)out_size;

  // 1) pack tgt & memory to planar, transpose W_alpha
  pack3_kernel<<<(NN * DD + 255) / 256, 256, 0, stream>>>(tgt, T_p, NN);
  pack3_kernel<<<(EE * DD + 255) / 256, 256, 0, stream>>>(memory, Mem_p, EE);
  transpose_walpha<<<(MULA * DD + 255) / 256, 256, 0, stream>>>(W_alpha, WaT);

  // 2) tv = T @ W_dst^T ; mv = Mem @ W_src^T
  gemm_wmma_f32<<<gemm_grid(3 * NN, DD), 256, 0, stream>>>(T_p, W_dst, tv_p, 3 * NN, DD, DD);
  gemm_wmma_f32<<<gemm_grid(3 * EE, DD), 256, 0, stream>>>(Mem_p, W_src, B1, 3 * EE, DD, DD);

  // 3) msg = mv + tv(broadcast)
  msg_add_kernel<<<(EE * DD + 255) / 256, 256, 0, stream>>>(B1, tv_p);

  // 4) edge MLP + geometric products (B1: msg -> p_v in place; p_sc out)
  edge_kernel<<<EE, 256, 0, stream>>>(edge_sc, edge_sh, fc_w1, fc_b1, fc_w2, fc_b2,
                                      fc_w3, fc_b3, B1, p_sc);

  // 5) attention logits: a_pre = p_sc @ W_alpha, LN + lrelu + head-dot, softmax over k
  gemm_wmma_f32<<<gemm_grid(EE, MULA), 256, 0, stream>>>(p_sc, WaT, a_pre, EE, MULA, DD);
  alpha_kernel<<<EE, 64, 0, stream>>>(a_pre, ln_a_g, ln_a_b, a_dot, logit);
  softmax_kernel<<<(NN * HH + 255) / 256, 256, 0, stream>>>(logit, aw);

  // 6) val = p_v @ W_value^T (reuse Mem_p), attention-weighted combine, proj
  gemm_wmma_f32<<<gemm_grid(3 * EE, DD), 256, 0, stream>>>(B1, W_value, Mem_p, 3 * EE, DD, DD);
  attn_combine<<<(NN * DD + 255) / 256, 256, 0, stream>>>(aw, Mem_p, out_p);
  gemm_wmma_f32<<<gemm_grid(3 * NN, DD), 256, 0, stream>>>(out_p, W_proj, proj_p, 3 * NN, DD, DD);

  // 7) x1 = vn_ln(tv + proj)
  vnln_kernel<<<NN, 256, 0, stream>>>(tv_p, proj_p, n1_g, n1_b, x1_p, 0);

  // 8) FF block: y = x1 @ W_ff1^T ; dvec = y @ W_ffd^T ; gate ; y2 = y @ W_ff2^T
  gemm_wmma_f32<<<gemm_grid(3 * NN, FFF), 256, 0, stream>>>(x1_p, W_ff1, y_p, 3 * NN, FFF, DD);
  gemm_wmma_f32<<<gemm_grid(3 * NN, FFF), 256, 0, stream>>>(y_p, W_ffd, dv_p, 3 * NN, FFF, FFF);
  gate_kernel<<<(NN * FFF + 255) / 256, 256, 0, stream>>>(y_p, dv_p);
  gemm_wmma_f32<<<gemm_grid(3 * NN, DD), 256, 0, stream>>>(y_p, W_ff2, y2_p, 3 * NN, DD, FFF);

  // 9) out = vn_ln(x1 + y2) interleaved to (N, 3D)
  vnln_kernel<<<NN, 256, 0, stream>>>(x1_p, y2_p, n2_g, n2_b, out, 1);
}